// VQEmbeddingEMA_67267777790570
// MI455X (gfx1250) — compile-verified
//
#include <hip/hip_runtime.h>
#include <hip/hip_bf16.h>

typedef float v2f __attribute__((ext_vector_type(2)));
typedef float v8f __attribute__((ext_vector_type(8)));

#define EMB_D 64
#define TILE_N 16
#define KCODES 1024
#define LDS_STRIDE 68   // 64 + 4 pad: bank = (4n + c) mod 64 -> conflict-free B reads

// ---------------------------------------------------------------------------
// Kernel 0: wsq[k] = ||W[k]||^2
// ---------------------------------------------------------------------------
__global__ __launch_bounds__(256) void vq_wsq_kernel(const float* __restrict__ W,
                                                     float* __restrict__ wsq, int K) {
  int k = blockIdx.x * blockDim.x + threadIdx.x;
  if (k >= K) return;
  const float4* w4 = (const float4*)(W + (size_t)k * EMB_D);
  float s = 0.f;
#pragma unroll
  for (int i = 0; i < EMB_D / 4; ++i) {
    float4 v = w4[i];
    s += v.x * v.x + v.y * v.y + v.z * v.z + v.w * v.w;
  }
  wsq[k] = s;
}

// ---------------------------------------------------------------------------
// Kernel 1: fused distance-matmul + argmin via V_WMMA_F32_16X16X4_F32.
// Whole codebook staged in LDS (276KB of the WGP's 320KB), padded stride 68
// so B-fragment ds_load_b64 reads are bank-conflict-free. One wave = 16
// tokens; block = 8 waves = 128 tokens sharing the LDS copy. Two N-tiles in
// flight per iteration -> two independent WMMA accumulation chains.
// dist[m][n] = ||w_n||^2 - 2 * (x_m . w_n)   (||x||^2 dropped: row-constant)
// ---------------------------------------------------------------------------
__global__ __launch_bounds__(256) void vq_argmin_kernel(
    const float* __restrict__ x, const float* __restrict__ W,
    const float* __restrict__ wsq_g, int* __restrict__ idx_out,
    unsigned int* __restrict__ counts, int K) {
  __shared__ float lds_w[KCODES * LDS_STRIDE];
  __shared__ float lds_wsq[KCODES];

  // Cooperative stage: W (row-padded) and wsq into LDS.
  for (int c = threadIdx.x; c < (KCODES * EMB_D) / 4; c += 256) {
    const int row  = c >> 4;          // 16 float4 per row
    const int col4 = (c & 15) << 2;
    float4 v = ((const float4*)W)[c];
    *(float4*)(lds_w + row * LDS_STRIDE + col4) = v;
  }
  for (int c = threadIdx.x; c < KCODES; c += 256) lds_wsq[c] = wsq_g[c];

  const int lane = threadIdx.x & 31;
  const int wave = threadIdx.x >> 5;
  const int tok0 = blockIdx.x * 128 + wave * 16;
  const int m    = lane & 15;          // row (token) within tile / col for B
  const int koff = (lane >> 4) << 1;   // upper half-wave holds K+2,K+3

  // Preload A fragments: 16 chunks of K=4 in native A layout (32 VGPRs).
  const float* xrow = x + (size_t)(tok0 + m) * EMB_D + koff;
  v2f a[16];
#pragma unroll
  for (int kk = 0; kk < 16; ++kk) a[kk] = *(const v2f*)(xrow + kk * 4);

  float bestD[8];
  int   bestI[8];
#pragma unroll
  for (int r = 0; r < 8; ++r) { bestD[r] = 3.4e38f; bestI[r] = 0; }

  __syncthreads();

  const int ntiles = KCODES / TILE_N;  // 64
  for (int nt = 0; nt < ntiles; nt += 2) {
    const int n0 = nt * TILE_N + m;
    const int n1 = n0 + TILE_N;
    const float* b0p = lds_w + n0 * LDS_STRIDE + koff;
    const float* b1p = lds_w + n1 * LDS_STRIDE + koff;

    v2f b0[16], b1[16];
#pragma unroll
    for (int kk = 0; kk < 16; ++kk) {
      b0[kk] = *(const v2f*)(b0p + kk * 4);
      b1[kk] = *(const v2f*)(b1p + kk * 4);
    }

    v8f acc0 = {};
    v8f acc1 = {};
#pragma unroll
    for (int kk = 0; kk < 16; ++kk) {
      acc0 = __builtin_amdgcn_wmma_f32_16x16x4_f32(
          false, a[kk], false, b0[kk], (short)0, acc0, false, false);
      acc1 = __builtin_amdgcn_wmma_f32_16x16x4_f32(
          false, a[kk], false, b1[kk], (short)0, acc1, false, false);
    }

    const float w0 = lds_wsq[n0];
    const float w1 = lds_wsq[n1];
#pragma unroll
    for (int r = 0; r < 8; ++r) {
      // C layout: VGPR r holds row r (lanes 0-15) / row r+8 (lanes 16-31).
      float d0 = fmaf(-2.0f, acc0[r], w0);
      if (d0 < bestD[r]) { bestD[r] = d0; bestI[r] = n0; }
      float d1 = fmaf(-2.0f, acc1[r], w1);
      if (d1 < bestD[r]) { bestD[r] = d1; bestI[r] = n1; }
    }
  }

  // Reduce argmin across the 16 lanes of each half-wave (masks stay in-half).
#pragma unroll
  for (int r = 0; r < 8; ++r) {
#pragma unroll
    for (int off = 8; off >= 1; off >>= 1) {
      float od = __shfl_xor(bestD[r], off, 32);
      int   oi = __shfl_xor(bestI[r], off, 32);
      if (od < bestD[r] || (od == bestD[r] && oi < bestI[r])) {
        bestD[r] = od; bestI[r] = oi;
      }
    }
  }

  // lane 0 -> rows 0..7, lane 16 -> rows 8..15
  if (m == 0) {
    const int rowbase = tok0 + (lane >> 4) * 8;
#pragma unroll
    for (int r = 0; r < 8; ++r) {
      idx_out[rowbase + r] = bestI[r];
      atomicAdd(&counts[bestI[r]], 1u);
    }
  }
}

// ---------------------------------------------------------------------------
// Kernel 2: quantized_st = x + (W[idx] - x); accumulate sum((q - x)^2).
// 16 threads per token (float4 per thread), block = 16 tokens.
// ---------------------------------------------------------------------------
__global__ __launch_bounds__(256) void vq_gather_kernel(
    const float* __restrict__ x, const float* __restrict__ W,
    const int* __restrict__ idx, float* __restrict__ out,
    float* __restrict__ sqsum) {
  __shared__ float red[256];
  const int t     = threadIdx.x;
  const int token = blockIdx.x * 16 + (t >> 4);
  const int chunk = t & 15;
  const int id    = idx[token];

  const float4* x4 = (const float4*)x;
  const float4* w4 = (const float4*)W;
  float4 xv = x4[(size_t)token * 16 + chunk];
  float4 wv = w4[(size_t)id * 16 + chunk];
  float4 dv = make_float4(wv.x - xv.x, wv.y - xv.y, wv.z - xv.z, wv.w - xv.w);
  float4 ov = make_float4(xv.x + dv.x, xv.y + dv.y, xv.z + dv.z, xv.w + dv.w);
  ((float4*)out)[(size_t)token * 16 + chunk] = ov;

  red[t] = dv.x * dv.x + dv.y * dv.y + dv.z * dv.z + dv.w * dv.w;
  __syncthreads();
#pragma unroll
  for (int off = 128; off > 0; off >>= 1) {
    if (t < off) red[t] += red[t + off];
    __syncthreads();
  }
  if (t == 0) atomicAdd(sqsum, red[0]);
}

// ---------------------------------------------------------------------------
// Kernel 3: loss = 3 * MSE; perplexity = exp(-sum p log(p+1e-10)); usage.
// ---------------------------------------------------------------------------
__global__ __launch_bounds__(1024) void vq_finalize_kernel(
    const unsigned int* __restrict__ counts, const float* __restrict__ sqsum,
    float* __restrict__ out3, int K, float inv_ntok, float inv_nelem) {
  __shared__ float redH[1024];
  __shared__ float redU[1024];
  const int t = threadIdx.x;
  float c = (t < K) ? (float)counts[t] : 0.f;
  float p = c * inv_ntok;
  redH[t] = p * logf(p + 1e-10f);
  redU[t] = (c >= 1.f) ? 1.f : 0.f;
  __syncthreads();
#pragma unroll
  for (int off = 512; off > 0; off >>= 1) {
    if (t < off) { redH[t] += redH[t + off]; redU[t] += redU[t + off]; }
    __syncthreads();
  }
  if (t == 0) {
    out3[0] = 3.0f * sqsum[0] * inv_nelem;  // q_latent + 2 * e_latent
    out3[1] = expf(-redH[0]);               // perplexity
    out3[2] = redU[0];                      // usage
  }
}

// ---------------------------------------------------------------------------
extern "C" void kernel_launch(void* const* d_in, const int* in_sizes, int n_in,
                              void* d_out, int out_size, void* d_ws, size_t ws_size,
                              hipStream_t stream) {
  const float* x = (const float*)d_in[0];
  const float* W = (const float*)d_in[1];
  const int n_elem = in_sizes[0];          // 32*4096*64 = 8388608
  const int K      = in_sizes[1] / EMB_D;  // 1024
  const int n_tok  = n_elem / EMB_D;       // 131072
  float* out = (float*)d_out;

  unsigned char* ws = (unsigned char*)d_ws;
  unsigned int* counts = (unsigned int*)ws;            // [0, 4KB)
  float*        sqsum  = (float*)(ws + 4096);          // 4 bytes
  float*        wsq    = (float*)(ws + 8192);          // [8KB, 12KB)
  int*          idx    = (int*)(ws + 12288);           // n_tok ints

  hipMemsetAsync(ws, 0, 4096 + 4, stream);             // counts + sqsum
  vq_wsq_kernel<<<(K + 255) / 256, 256, 0, stream>>>(W, wsq, K);
  vq_argmin_kernel<<<n_tok / 128, 256, 0, stream>>>(x, W, wsq, idx, counts, K);
  vq_gather_kernel<<<n_tok / 16, 256, 0, stream>>>(x, W, idx, out, sqsum);
  vq_finalize_kernel<<<1, 1024, 0, stream>>>(counts, sqsum, out + n_elem, K,
                                             1.0f / (float)n_tok,
                                             1.0f / (float)n_elem);
}